// DPS_topk_9088150798854
// MI455X (gfx1250) — compile-verified
//
#include <hip/hip_runtime.h>
#include <cstdint>
#include <cstddef>

typedef float v8f __attribute__((ext_vector_type(8)));
typedef float v2f __attribute__((ext_vector_type(2)));
typedef unsigned int u32x4 __attribute__((ext_vector_type(4)));
typedef int i32x4 __attribute__((ext_vector_type(4)));
typedef int i32x8 __attribute__((ext_vector_type(8)));

#define BSZ      8
#define MROW     64
#define NCOL     4096
#define KSEL     32
#define TINV     0.2f        // 1/TEMP, TEMP = 5
#define EPSF     1e-20f
#define NTHREADS 256
#define PER_THR  (NCOL / NTHREADS)   // 16

#if defined(__has_builtin)
#if __has_builtin(__builtin_amdgcn_tensor_load_to_lds)
#define HAVE_TDM 1
#endif
#endif

#ifdef HAVE_TDM
// Issue one TDM 1-D row copy: NCOL fp32 from global `src` into LDS at byte
// offset `lds_off`.  D# layout per CDNA5 ISA §8 (group0 128b, group1 256b).
__device__ __forceinline__ void tdm_load_row(const float* src, unsigned lds_off) {
    const unsigned long long ga = (unsigned long long)(uintptr_t)src;

    u32x4 g0;
    g0.x = 1u;                                   // count=1, user mode, no gather
    g0.y = lds_off;                              // lds_addr (bytes)
    g0.z = (unsigned)(ga & 0xFFFFFFFFull);       // global_addr[31:0]
    g0.w = (unsigned)((ga >> 32) & 0x01FFFFFFull) | (2u << 30);  // addr[56:32] | type=2

    i32x8 g1;
    g1[0] = (int)(2u << 16);                     // data_size=2 (4B); mask/flags/pad = 0
    g1[1] = (int)((NCOL & 0xFFFFu) << 16);       // tensor_dim0[15:0] @ bits 63:48
    g1[2] = (int)(1u << 16);                     // tensor_dim0 hi = 0; tensor_dim1 = 1
    g1[3] = (int)((NCOL & 0xFFFFu) << 16);       // tensor_dim1 hi = 0; tile_dim0 = NCOL
    g1[4] = 1;                                   // tile_dim1 = 1; tile_dim2 = 0
    g1[5] = NCOL;                                // tensor_dim0_stride[31:0]
    g1[6] = (int)((NCOL & 0xFFFFu) << 16);       // stride hi = 0; tensor_dim1_stride[15:0]
    g1[7] = 0;                                   // tensor_dim1_stride hi

    i32x4 gz = {0, 0, 0, 0};                     // groups 2/3 unused (<=2D tensor)

#if __has_include(<hip/amd_detail/amd_gfx1250_TDM.h>)
    i32x8 gz8 = {0, 0, 0, 0, 0, 0, 0, 0};
    __builtin_amdgcn_tensor_load_to_lds(g0, g1, gz, gz, gz8, 0);   // 6-arg toolchain
#else
    __builtin_amdgcn_tensor_load_to_lds(g0, g1, gz, gz, 0);        // 5-arg toolchain
#endif
}
#endif

__global__ __launch_bounds__(NTHREADS)
void dps_topk_kernel(const float* __restrict__ logits,
                     const float* __restrict__ u,
                     float* __restrict__ out)
{
    __shared__ float              s_work[NCOL];   // TDM: logits row -> overwritten w/ perturbed
    __shared__ float              s_p[NCOL];      // TDM: u row      -> overwritten w/ exp(./T)
    __shared__ unsigned char      s_rank[NCOL];   // rank among sorted selected (32 = unselected)
    __shared__ unsigned long long s_red[NTHREADS];
    __shared__ float              s_fred[NTHREADS];
    __shared__ int                s_sel[KSEL];
    __shared__ int                s_sorted[KSEL];
    __shared__ float              s_invD[KSEL];

    const int tid = threadIdx.x;
    const int bm  = blockIdx.x;            // 0 .. BSZ*MROW-1
    const int m   = bm & (MROW - 1);

    const float* lrow = logits + (size_t)m  * NCOL;
    const float* urow = u      + (size_t)bm * NCOL;
    float*       orow = out    + (size_t)bm * (size_t)KSEL * NCOL;

    const float wm = __expf(__logf(EPSF) * TINV);   // masked softmax weight ~9.999e-5

    // ---------- Phase 0: stage input rows into LDS via the Tensor Data Mover ----------
#ifdef HAVE_TDM
    if (tid == 0) {      // one wave issues the DMA; EXEC is ignored by tensor ops
        tdm_load_row(lrow, (unsigned)(uintptr_t)(void*)s_work);
        tdm_load_row(urow, (unsigned)(uintptr_t)(void*)s_p);
    }
    __builtin_amdgcn_s_wait_tensorcnt(0);   // wave-local TENSORcnt drain (s_wait_tensorcnt 0x0)
    __syncthreads();                        // publish LDS to the other 7 waves
#endif

    // ---------- Phase 1: gumbel-perturb, exp, (re)stage into LDS ----------
    float tsum = 0.0f;
    #pragma unroll
    for (int j = 0; j < PER_THR; ++j) {
        const int n = j * NTHREADS + tid;
#ifdef HAVE_TDM
        const float lg = s_work[n];
        const float uu = s_p[n];
#else
        if (j + 1 < PER_THR) {
            __builtin_prefetch(&urow[(j + 1) * NTHREADS + tid], 0, 0);  // global_prefetch_b8
            __builtin_prefetch(&lrow[(j + 1) * NTHREADS + tid], 0, 0);
        }
        const float lg = lrow[n];
        const float uu = __builtin_nontemporal_load(&urow[n]);
#endif
        const float gn = -0.001f * __logf(-__logf(uu + EPSF) + EPSF);
        const float pert = lg + gn;
        const float pv = __expf(pert * TINV);
        s_work[n] = pert;          // same-thread overwrite of staged slot: no hazard
        s_p[n]    = pv;
        s_rank[n] = (unsigned char)KSEL;
        tsum += pv;
    }
    s_fred[tid] = tsum;
    __syncthreads();

    // total sum S of p[n]  (barrier-based tree reduce; no wave-width assumptions)
    for (int off = NTHREADS / 2; off > 0; off >>= 1) {
        if (tid < off) s_fred[tid] += s_fred[tid + off];
        __syncthreads();
    }

    // ---------- Phase 2: top-32 by perturbed value (ties -> lower index) ----------
    for (int it = 0; it < KSEL; ++it) {
        unsigned long long best = 0ull;
        #pragma unroll
        for (int j = 0; j < PER_THR; ++j) {
            const int n = j * NTHREADS + tid;
            const unsigned int bits = __float_as_uint(s_work[n]);
            const unsigned int mono = bits ^ ((bits & 0x80000000u) ? 0xFFFFFFFFu : 0x80000000u);
            const unsigned long long key =
                ((unsigned long long)mono << 32) | (unsigned int)(NCOL - 1 - n);
            best = (key > best) ? key : best;
        }
        s_red[tid] = best;
        __syncthreads();
        for (int off = NTHREADS / 2; off > 0; off >>= 1) {
            if (tid < off) {
                const unsigned long long o = s_red[tid + off];
                if (o > s_red[tid]) s_red[tid] = o;
            }
            __syncthreads();
        }
        if (tid == 0) {
            const int nsel = (NCOL - 1) - (int)(s_red[0] & 0xFFFFFFFFull);
            s_sel[it] = nsel;
            s_work[nsel] = -__builtin_inff();   // remove from next rounds
        }
        __syncthreads();
    }

    // sort the 32 selected indices ascending (rank sort), build rank table
    if (tid < KSEL) {
        const int v = s_sel[tid];
        int pos = 0;
        #pragma unroll
        for (int i = 0; i < KSEL; ++i) pos += (s_sel[i] < v) ? 1 : 0;
        s_sorted[pos] = v;
    }
    __syncthreads();
    if (tid < KSEL) s_rank[s_sorted[tid]] = (unsigned char)tid;
    __syncthreads();

    // incremental denominators: D_k = S - (1-wm) * sum_{j<k} p[sorted_j]
    if (tid == 0) {
        float cur = s_fred[0];
        for (int k = 0; k < KSEL; ++k) {
            s_invD[k] = 1.0f / cur;
            cur -= (1.0f - wm) * s_p[s_sorted[k]];
        }
    }
    __syncthreads();

    // ---------- Phase 3: soft = invD (32) outer-product p (4096) on the matrix pipe
    //            (V_WMMA_F32_16X16X4_F32), VALU mask/hard fixups, NT streaming stores ----
    const int lane  = tid & 31;
    const int wave  = tid >> 5;
    const int lhalf = lane >> 4;     // 0: lanes 0-15, 1: lanes 16-31
    const int lmod  = lane & 15;

    for (int t = wave; t < NCOL / 16; t += NTHREADS / 32) {   // uniform per wave -> EXEC all 1s
        const int   n  = t * 16 + lmod;
        const float pn = s_p[n];
        const int   r  = (int)s_rank[n];

        v2f bmat;                                  // B 4x16: K=0 row = p tile, rest 0
        bmat.x = (lhalf == 0) ? pn : 0.0f;
        bmat.y = 0.0f;

        #pragma unroll
        for (int kt = 0; kt < 2; ++kt) {
            const int kbase = kt * 16;
            v2f amat;                              // A 16x4: K=0 column = invD, rest 0
            amat.x = (lhalf == 0) ? s_invD[kbase + lmod] : 0.0f;
            amat.y = 0.0f;

            v8f c = {};
            // D(16x16) = A(16x4) x B(4x16) + 0  -> c[m][n] = invD[kbase+m] * p[n]
            c = __builtin_amdgcn_wmma_f32_16x16x4_f32(
                    false, amat, false, bmat, (short)0, c, false, false);

            #pragma unroll
            for (int i = 0; i < 8; ++i) {
                const int   k    = kbase + i + (lhalf << 3);     // C/D layout: VGPR i -> M=i / M=i+8
                const float soft = c[i] * ((r < k) ? wm : 1.0f);
                const float hard = (r == k) ? 1.0f : 0.0f;
                const float val  = (hard - soft) + soft;          // straight-through, as in reference
                __builtin_nontemporal_store(val, &orow[(size_t)k * NCOL + n]);
            }
        }
    }
}

extern "C" void kernel_launch(void* const* d_in, const int* in_sizes, int n_in,
                              void* d_out, int out_size, void* d_ws, size_t ws_size,
                              hipStream_t stream) {
    const float* logits = (const float*)d_in[0];   // (M, N) fp32
    const float* u      = (const float*)d_in[1];   // (BS, M, N) fp32
    float*       out    = (float*)d_out;           // (BS, M, K, N) fp32

    dim3 grid(BSZ * MROW);     // 512 workgroups, one (b,m) row each
    dim3 block(NTHREADS);      // 8 wave32s
    hipLaunchKernelGGL(dps_topk_kernel, grid, block, 0, stream, logits, u, out);
}